// PoSEmbedding_51067161149885
// MI455X (gfx1250) — compile-verified
//
#include <hip/hip_runtime.h>
#include <stdint.h>

// Problem constants (fixed by the reference): V=64, D=512, K=4.
#define KIDS 4
#define DIM  512
#define DV   (DIM / 4)      // 128 float4 per token
#define VOCAB 64
#define TPB  256            // 8 waves of 32 (wave32)
#define TOKS_PER_BLOCK 256

typedef __attribute__((ext_vector_type(4))) unsigned int u32x4;
typedef __attribute__((ext_vector_type(8))) int          i32x8;
typedef __attribute__((ext_vector_type(4))) int          i32x4;
typedef __attribute__((ext_vector_type(4))) float        f32x4;

__global__ __launch_bounds__(TPB)
void PoSEmbedding_bag_kernel(const int* __restrict__ ids,     // [ntok, 4] int32
                             const float* __restrict__ table, // [64, 512] f32
                             float* __restrict__ out,         // [ntok, 512] f32
                             int ntok)
{
    // 64 x 512 f32 = 128 KB staged table (<= 320 KB/WGP on CDNA5)
    __shared__ float lds_table[VOCAB * DIM];

#if __has_builtin(__builtin_amdgcn_tensor_load_to_lds)
    // ---- Stage rows 1..63 via the Tensor Data Mover (one DMA descriptor) ----
    if (threadIdx.x == 0) {
        uint64_t gaddr = (uint64_t)(uintptr_t)(table + DIM);          // skip row 0
        uint32_t laddr = (uint32_t)(uintptr_t)(&lds_table[DIM]);      // low 32b of LDS flat addr = LDS byte offset

        u32x4 g0;
        g0[0] = 1u;                                                    // count=1, user descriptor
        g0[1] = laddr;                                                 // lds_addr [63:32]
        g0[2] = (uint32_t)(gaddr & 0xFFFFFFFFu);                       // global_addr lo
        g0[3] = (uint32_t)((gaddr >> 32) & 0x1FFFFFFu) | (2u << 30);   // global_addr hi | type=2

        i32x8 g1;
        g1[0] = (int)(2u << 16);                         // data_size=4B; mask/flags=0
        g1[1] = (int)((uint32_t)(DIM & 0xFFFF) << 16);   // tensor_dim0[15:0]=512 @ bits[63:48]
        g1[2] = (int)((uint32_t)(VOCAB - 1) << 16);      // tensor_dim1[15:0]=63  @ bits[95:80]
        g1[3] = (int)((uint32_t)(DIM & 0xFFFF) << 16);   // tile_dim0=512         @ bits[127:112]
        g1[4] = (int)(VOCAB - 1);                        // tile_dim1=63, tile_dim2=0
        g1[5] = DIM;                                     // tensor_dim0_stride=512 (elements)
        g1[6] = 0;                                       // stride hi / dim1_stride lo
        g1[7] = 0;                                       // dim1_stride hi

        i32x4 gz = {0, 0, 0, 0};
#if defined(__clang_major__) && (__clang_major__ >= 23)
        i32x8 gz8 = {0, 0, 0, 0, 0, 0, 0, 0};
        __builtin_amdgcn_tensor_load_to_lds(g0, g1, gz, gz, gz8, 0);
#else
        __builtin_amdgcn_tensor_load_to_lds(g0, g1, gz, gz, 0);
#endif
        __builtin_amdgcn_s_wait_tensorcnt((short)0);     // wave 0 blocks until TDM done
    }
#else
    // Fallback: cooperative vectorized copy of rows 1..63
    {
        const f32x4* src = (const f32x4*)(table + DIM);
        f32x4* dst = (f32x4*)(&lds_table[DIM]);
        for (int i = threadIdx.x; i < (VOCAB - 1) * DV; i += TPB) dst[i] = src[i];
    }
#endif

    // Padding row: nn.Embedding(padding_idx=0) embeds id 0 to the zero vector.
    // 256 threads x 2 floats cover the 512-float row 0 (disjoint from the TDM tile).
    lds_table[2 * threadIdx.x]     = 0.0f;
    lds_table[2 * threadIdx.x + 1] = 0.0f;

    __syncthreads();   // TDM data + zero row visible to all 8 waves

    // ---- Gather + sum: one token per wave; lanes stride over D in float4 ----
    const int lane = threadIdx.x & 31;
    const int wave = threadIdx.x >> 5;
    const int tok_base = blockIdx.x * TOKS_PER_BLOCK;

    const f32x4* lt = (const f32x4*)lds_table;           // [VOCAB][DV]

    for (int i = wave; i < TOKS_PER_BLOCK; i += TPB / 32) {
        int tok = tok_base + i;
        if (tok >= ntok) break;
        // token index is wave-uniform: force scalar so the 4 ids load via SMEM
        tok = __builtin_amdgcn_readfirstlane(tok);

        const i32x4 id4 = *((const i32x4*)ids + tok);
        const int r0 = id4.x * DV;
        const int r1 = id4.y * DV;
        const int r2 = id4.z * DV;
        const int r3 = id4.w * DV;

        f32x4* od = (f32x4*)out + (size_t)tok * DV;

#pragma unroll
        for (int j = 0; j < 4; ++j) {
            const int dv = lane + j * 32;                // conflict-free ds_load_b128
            f32x4 a = lt[r0 + dv];
            f32x4 b = lt[r1 + dv];
            f32x4 c = lt[r2 + dv];
            f32x4 d = lt[r3 + dv];
            f32x4 s = (a + b) + (c + d);
            // write-once stream: nontemporal b128 store, don't pollute caches
            __builtin_nontemporal_store(s, od + dv);
        }
    }
}

extern "C" void kernel_launch(void* const* d_in, const int* in_sizes, int n_in,
                              void* d_out, int out_size, void* d_ws, size_t ws_size,
                              hipStream_t stream) {
    const int*   ids   = (const int*)d_in[0];    // pos_ids  [B,T,K] int32
    const float* table = (const float*)d_in[1];  // emb_table [V,D] f32
    float*       out   = (float*)d_out;          // [B,T,D] f32

    const int ntok   = in_sizes[0] / KIDS;       // B*T = 65536
    const int blocks = (ntok + TOKS_PER_BLOCK - 1) / TOKS_PER_BLOCK;

    hipLaunchKernelGGL(PoSEmbedding_bag_kernel, dim3(blocks), dim3(TPB), 0, stream,
                       ids, table, out, ntok);
}